// PromptedAttention_69131793596922
// MI455X (gfx1250) — compile-verified
//
#include <hip/hip_runtime.h>
#include <math.h>

typedef __attribute__((ext_vector_type(2))) float v2f;
typedef __attribute__((ext_vector_type(4))) float v4f;
typedef __attribute__((ext_vector_type(8))) float v8f;
typedef __attribute__((ext_vector_type(4))) int   v4i;

#define B_    8
#define Cc    768
#define NH    12
#define HD    64
#define NT    8
#define Nn    1032          // H*W + NT
#define BN    (B_*Nn)       // 8256 token rows
#define BHh   (B_*NH)       // 96 attention batches
#define C3    (3*Cc)        // 2304
#define NPIX  1024          // H*W
#define SCALE 0.125f        // hd^-0.5
#define NKT   65            // ceil(1032/16) key tiles
#define SP    1040          // padded score-row length (65*16)
#define CHUNK 128           // K-chunk for double-buffered GEMM pipeline

// ---- CDNA5 async global->LDS staging (guarded; falls back to plain loads) --
#if defined(__has_builtin)
#if __has_builtin(__builtin_amdgcn_global_load_async_to_lds_b128)
#define HAVE_ASYNC_LDS 1
#endif
#endif
#ifndef HAVE_ASYNC_LDS
#define HAVE_ASYNC_LDS 0
#endif

__device__ __forceinline__ void copy16_g2l(const float* gp, float* lp) {
#if HAVE_ASYNC_LDS
  __builtin_amdgcn_global_load_async_to_lds_b128(
      (__attribute__((address_space(1))) v4i*)gp,
      (__attribute__((address_space(3))) v4i*)lp, 0, 0);
#else
  *(v4f*)lp = *(const v4f*)gp;
#endif
}

__device__ __forceinline__ void async_wait_all() {
#if HAVE_ASYNC_LDS
#if __has_builtin(__builtin_amdgcn_s_wait_asynccnt)
  __builtin_amdgcn_s_wait_asynccnt(0);
#else
  asm volatile("s_wait_asynccnt 0x0" ::: "memory");
#endif
#endif
}

__device__ __forceinline__ v8f wmma4(v2f a, v2f b, v8f c) {
  // D = A(16x4 f32) * B(4x16 f32) + C(16x16 f32)
  return __builtin_amdgcn_wmma_f32_16x16x4_f32(false, a, false, b, (short)0, c,
                                               false, false);
}

// ---------------------------------------------------------------------------
// Unified GEMM: 32x128 block tile, 4 waves, each wave computes a 2x2 grid of
// 16x16 WMMA tiles.  A tile staged through LDS with a double-buffered async
// global->LDS pipeline (ASYNCcnt).  B read from global (L2-resident weights).
// MODE 0: qkv  = concat(vp,x) @ w + bias -> scatter to q/k/v [bh][n][hd]
// MODE 1: proj = ain @ w + bias          -> scatter to (x_out, vp_out)
// ---------------------------------------------------------------------------
template<int LDB, int MODE>
__global__ __launch_bounds__(128)
void gemm_kernel(const float* __restrict__ ain, const float* __restrict__ vp,
                 const float* __restrict__ w, const float* __restrict__ bias,
                 float* __restrict__ o0, float* __restrict__ o1,
                 float* __restrict__ o2)
{
  __shared__ __align__(16) float As[2][32 * CHUNK];   // 2 x 16 KB
  const int rt  = blockIdx.x;          // 32-row tile
  const int cg  = blockIdx.y;          // 128-col group
  const int tid = threadIdx.x;
  const int wave = tid >> 5, lane = tid & 31;
  const int half = lane >> 4, lrow = lane & 15;
  const int oc0  = cg * 128 + wave * 32;

  auto stage = [&](int ck, int bb) {
    for (int it = 0; it < 8; ++it) {
      int idx = it * 128 + tid;        // 1024 x 16B = 32x128 floats
      int row = idx >> 5;
      int c4  = (idx & 31) << 2;
      int t   = rt * 32 + row;
      const float* gp;
      if (MODE == 0) {
        int b = t / Nn, n = t % Nn;
        gp = (n < NT) ? vp + ((size_t)b * NT + n) * Cc + ck + c4
                      : ain + ((size_t)b * NPIX + (n - NT)) * Cc + ck + c4;
      } else {
        gp = ain + (size_t)t * Cc + ck + c4;
      }
      copy16_g2l(gp, &As[bb][row * CHUNK + c4]);
    }
  };

  stage(0, 0);
  async_wait_all();
  __syncthreads();

  v8f a00 = {}, a01 = {}, a10 = {}, a11 = {};
  for (int ck = 0; ck < Cc; ck += CHUNK) {
    const int cur = (ck / CHUNK) & 1;
    if (ck + CHUNK < Cc) stage(ck + CHUNK, cur ^ 1);   // prefetch next chunk
    const float* Ab = &As[cur][0];
    for (int c = 0; c < CHUNK; c += 4) {
      v2f a0 = *(const v2f*)(Ab + lrow * CHUNK + c + 2 * half);
      v2f a1 = *(const v2f*)(Ab + (lrow + 16) * CHUNK + c + 2 * half);
      const size_t gr0 = (size_t)(ck + c + 2 * half) * LDB + oc0;
      v2f b0, b1;
      b0.x = w[gr0 + lrow];
      b0.y = w[gr0 + LDB + lrow];
      b1.x = w[gr0 + 16 + lrow];
      b1.y = w[gr0 + LDB + 16 + lrow];
      a00 = wmma4(a0, b0, a00);
      a01 = wmma4(a0, b1, a01);
      a10 = wmma4(a1, b0, a10);
      a11 = wmma4(a1, b1, a11);
    }
    async_wait_all();
    __syncthreads();
  }

  // epilogue: bias + scatter
  for (int mi = 0; mi < 2; ++mi) {
    for (int ni = 0; ni < 2; ++ni) {
      v8f acc = (mi == 0) ? (ni == 0 ? a00 : a01) : (ni == 0 ? a10 : a11);
      const int oc = oc0 + ni * 16 + lrow;
      const float bval = bias[oc];
      for (int r = 0; r < 8; ++r) {
        const int t = rt * 32 + mi * 16 + r + 8 * half;
        const int b = t / Nn, n = t % Nn;
        const float val = acc[r] + bval;
        if (MODE == 0) {
          const int which = oc / Cc;
          const int rem   = oc % Cc;
          const int h = rem / HD, d = rem % HD;
          float* dst = (which == 0) ? o0 : (which == 1) ? o1 : o2;
          dst[(((size_t)b * NH + h) * Nn + n) * HD + d] = val;
        } else {
          if (n >= NT)
            o0[((size_t)b * NPIX + (n - NT)) * Cc + oc] = val;           // x_out
          else
            o0[(size_t)B_ * NPIX * Cc + ((size_t)b * NT + n) * Cc + oc] = val; // vp_out
        }
      }
    }
  }
}

// ---------------------------------------------------------------------------
// Decomposed relative-position terms.
// rel_h[bh][p][kh] = dot(q[bh][nt+p], rel_pos_h[qh-kh+31]); rel_w analogous.
// ---------------------------------------------------------------------------
__global__ __launch_bounds__(32)
void relpos_kernel(const float* __restrict__ q,
                   const float* __restrict__ rph, const float* __restrict__ rpw,
                   float* __restrict__ rel_h, float* __restrict__ rel_w)
{
  const int bh = blockIdx.x;
  const int p  = blockIdx.y;
  const int lane = threadIdx.x;
  const int qh = p >> 5, qw = p & 31;
  const float* qv = q + ((size_t)bh * Nn + NT + p) * HD;
  const float* rh = rph + (size_t)(qh - lane + 31) * HD;
  const float* rw = rpw + (size_t)(qw - lane + 31) * HD;
  float sh = 0.f, sw = 0.f;
  for (int c = 0; c < HD; ++c) {
    float qc = qv[c];
    sh += qc * rh[c];
    sw += qc * rw[c];
  }
  rel_h[((size_t)bh * NPIX + p) * 32 + lane] = sh;
  rel_w[((size_t)bh * NPIX + p) * 32 + lane] = sw;
}

// ---------------------------------------------------------------------------
// Fused attention for one (bh, 16-query tile).  All boundary handling via
// index clamping (no EXEC-masked loads in the WMMA loops): garbage from
// clamped addresses only lands in score columns that are overwritten with
// -1e30, or is multiplied by exactly-zero probabilities.
// Grid (96, 65), 128 threads (4 waves).
// ---------------------------------------------------------------------------
__global__ __launch_bounds__(128)
void attn_kernel(const float* __restrict__ q, const float* __restrict__ k,
                 const float* __restrict__ v,
                 const float* __restrict__ rel_h, const float* __restrict__ rel_w,
                 float* __restrict__ out)
{
  extern __shared__ float smem[];
  float* Qs      = smem;                   // 16*64
  float* S       = Qs + 16 * HD;           // 16*1040
  float* red     = S + 16 * SP;            // 16*8
  float* rowstat = red + 16 * 8;           // 16 max + 16 sum

  const int bh = blockIdx.x;
  const int qt = blockIdx.y;
  const int qbase = qt * 16;
  const int tid = threadIdx.x;
  const int wave = tid >> 5, lane = tid & 31;
  const int half = lane >> 4, lrow = lane & 15;

  // stage scaled Q tile; clamp row index (garbage rows never stored)
  for (int i = tid; i < 16 * HD; i += 128) {
    int r = i >> 6, c = i & 63;
    int qrow = qbase + r;
    int qc = (qrow < Nn) ? qrow : (Nn - 1);
    Qs[i] = q[((size_t)bh * Nn + qc) * HD + c] * SCALE;
  }
  __syncthreads();

  // ---- phase 1: S = (scale*Q) K^T + bias/mask ----
  for (int kt = wave; kt < NKT; kt += 4) {
    const int kb  = kt * 16 + lrow;
    const int kbc = (kb < Nn) ? kb : (Nn - 1);       // clamp, not mask
    const float* krow = k + ((size_t)bh * Nn + kbc) * HD;
    v8f acc = {};
    for (int c = 0; c < HD; c += 4) {
      v2f a = *(const v2f*)(Qs + lrow * HD + c + 2 * half);
      v2f b = *(const v2f*)(krow + c + 2 * half);
      acc = wmma4(a, b, acc);
    }
    const int kcol = kt * 16 + lrow;
    for (int r = 0; r < 8; ++r) {
      const int M = r + 8 * half;
      const int qrow = qbase + M;
      float sv = acc[r];
      if (kcol >= Nn) {
        sv = -1e30f;                       // padded keys -> prob 0
      } else if (qrow >= NT && qrow < Nn) {
        if (kcol >= NT) {                  // pixel-pixel: add rel-pos bias
          const int p = qrow - NT, pk = kcol - NT;
          sv += rel_h[((size_t)bh * NPIX + p) * 32 + (pk >> 5)]
              + rel_w[((size_t)bh * NPIX + p) * 32 + (pk & 31)];
        } else {
          sv -= 100.0f;                    // pixel query -> prompt key mask
        }
      }
      S[M * SP + kcol] = sv;
    }
  }
  __syncthreads();

  // ---- phase 2: softmax over each 1040-wide row (8 threads per row) ----
  {
    const int row = tid >> 3;
    const int sub = tid & 7;
    float m = -3.4e38f;
    for (int c = sub; c < SP; c += 8) m = fmaxf(m, S[row * SP + c]);
    red[row * 8 + sub] = m;
    __syncthreads();
    if (sub == 0) {
      float mm = red[row * 8];
      for (int i = 1; i < 8; ++i) mm = fmaxf(mm, red[row * 8 + i]);
      rowstat[row] = mm;
    }
    __syncthreads();
    const float mm = rowstat[row];
    float s = 0.f;
    for (int c = sub; c < SP; c += 8) {
      float e = __expf(S[row * SP + c] - mm);
      S[row * SP + c] = e;
      s += e;
    }
    red[row * 8 + sub] = s;
    __syncthreads();
    if (sub == 0) {
      float ss = 0.f;
      for (int i = 0; i < 8; ++i) ss += red[row * 8 + i];
      rowstat[16 + row] = ss;
    }
    __syncthreads();
    const float inv = 1.0f / rowstat[16 + row];
    for (int c = sub; c < SP; c += 8) S[row * SP + c] *= inv;
  }
  __syncthreads();

  // ---- phase 3: out = P @ V ; wave owns 16-wide hd chunk ----
  const int hc = wave * 16;
  const float* vbase = v + (size_t)bh * Nn * HD + hc + lrow;
  v8f acc = {};
  // main body: no boundary possible (kb1 <= 1027 < Nn)
  for (int ks = 0; ks < 1028; ks += 4) {
    const int kb0 = ks + 2 * half;
    v2f a = *(const v2f*)(S + lrow * SP + kb0);
    v2f b;
    b.x = vbase[(size_t)kb0 * HD];
    b.y = vbase[(size_t)(kb0 + 1) * HD];
    acc = wmma4(a, b, acc);
  }
  // tail: clamp V row index; P[:,K>=Nn] == 0 so clamped rows contribute 0
  for (int ks = 1028; ks < SP; ks += 4) {
    const int kb0 = ks + 2 * half;
    const int kc0 = (kb0 < Nn) ? kb0 : (Nn - 1);
    const int kc1 = (kb0 + 1 < Nn) ? (kb0 + 1) : (Nn - 1);
    v2f a = *(const v2f*)(S + lrow * SP + kb0);
    v2f b;
    b.x = vbase[(size_t)kc0 * HD];
    b.y = vbase[(size_t)kc1 * HD];
    acc = wmma4(a, b, acc);
  }
  const int b = bh / NH, h = bh % NH;
  for (int r = 0; r < 8; ++r) {
    const int M = r + 8 * half;
    const int qrow = qbase + M;
    if (qrow < Nn)
      out[((size_t)b * Nn + qrow) * Cc + h * HD + hc + lrow] = acc[r];
  }
}

// ---------------------------------------------------------------------------
extern "C" void kernel_launch(void* const* d_in, const int* in_sizes, int n_in,
                              void* d_out, int out_size, void* d_ws, size_t ws_size,
                              hipStream_t stream)
{
  const float* x     = (const float*)d_in[0];
  const float* vp    = (const float*)d_in[1];
  const float* qkvw  = (const float*)d_in[2];
  const float* qkvb  = (const float*)d_in[3];
  const float* projw = (const float*)d_in[4];
  const float* projb = (const float*)d_in[5];
  const float* rph   = (const float*)d_in[6];
  const float* rpw   = (const float*)d_in[7];
  (void)in_sizes; (void)n_in; (void)out_size; (void)ws_size;

  float* ws = (float*)d_ws;
  const size_t qkv_sz = (size_t)BHh * Nn * HD;    // 6,340,608 floats each
  const size_t rel_sz = (size_t)BHh * NPIX * 32;  // 3,145,728 floats each
  float* q  = ws;
  float* k  = q + qkv_sz;
  float* v  = k + qkv_sz;
  float* rh = v + qkv_sz;
  float* rw = rh + rel_sz;
  float* ao = rw + rel_sz;                        // BN*Cc attention output

  gemm_kernel<C3, 0><<<dim3(BN / 32, C3 / 128), 128, 0, stream>>>(
      x, vp, qkvw, qkvb, q, k, v);
  relpos_kernel<<<dim3(BHh, NPIX), 32, 0, stream>>>(q, rph, rpw, rh, rw);
  const size_t smem_bytes = (size_t)(16 * HD + 16 * SP + 16 * 8 + 32) * sizeof(float);
  attn_kernel<<<dim3(BHh, NKT), 128, smem_bytes, stream>>>(q, k, v, rh, rw, ao);
  gemm_kernel<Cc, 1><<<dim3(BN / 32, Cc / 128), 128, 0, stream>>>(
      ao, nullptr, projw, projb, (float*)d_out, nullptr, nullptr);
}